// PGMNOV2_55886114455572
// MI455X (gfx1250) — compile-verified
//
#include <hip/hip_runtime.h>
#include <hip/hip_bf16.h>
#include <math.h>

// ---------------------------------------------------------------------------
// Mamba neural-operator time stepper for MI455X (gfx1250, wave32, WMMA).
//
// Roofline: ~25 GFLOP total, ~65 MB live state (L2-resident at 192 MB) ->
// latency-bound. fp32 end-to-end; all GEMMs use V_WMMA_F32_16X16X4_F32.
// GEMM shapes are compile-time template constants so every fragment load
// becomes an immediate-offset global_load from a fixed base pointer
// (no per-step 64-bit address VALU), fully unrolled over K.
// ---------------------------------------------------------------------------

typedef __attribute__((ext_vector_type(2))) float v2f;
typedef __attribute__((ext_vector_type(8))) float v8f;

#define D_MODEL   256
#define D_INNER   512
#define D_STATE   16
#define DT_RANK   16
#define D_CONV    4
#define NLAYERS   4
#define NSEQ      12          // BATCH * K_STEPS
#define NGRID     512
#define NTOK      (NSEQ * NGRID)   // 6144
#define DT_STEP   0.01f

__device__ __forceinline__ float pgm_silu(float x)  { return x / (1.0f + __expf(-x)); }
__device__ __forceinline__ float pgm_gelu(float x)  { return 0.5f * x * (1.0f + erff(x * 0.70710678118654752f)); }
__device__ __forceinline__ float pgm_softplus(float x) { return (x > 20.0f) ? x : log1pf(expf(x)); }

// ---------------------------------------------------------------------------
// fp32 WMMA GEMM: C[M,N] = A[M,K] @ B[K,N] (+ epilogue).
// Block = 256 threads = 8 wave32s; each wave owns one 16x16 C tile;
// block tile = 128 (M) x 16 (N). Grid = (N/16, M/128). K % 4 == 0.
// All strides / K / epilogue are template constants -> immediate-offset
// vector loads, s_clause batching, fully unrolled WMMA stream.
// MODE 0: store                MODE 1: +bias, store
// MODE 2: +bias, softplus      MODE 3: C = clip(C + scale*acc, -1, 1)
// ---------------------------------------------------------------------------
template<int K, int LDA, int LDB, int LDC, int MODE>
__global__ __launch_bounds__(256)
void pgm_gemm16(const float* __restrict__ A,
                const float* __restrict__ B,
                float* __restrict__ C,
                const float* __restrict__ bias, float scale)
{
    const int lane  = threadIdx.x & 31;
    const int wave  = threadIdx.x >> 5;        // 0..7
    const int half  = lane >> 4;               // 0 or 1
    const int l     = lane & 15;
    const int mBase = blockIdx.y * 128 + wave * 16;
    const int nBase = blockIdx.x * 16;

    // Fixed base pointers; all K-step accesses are compile-time offsets.
    const float* __restrict__ Ap = A + (size_t)(mBase + l) * LDA + 2 * half;
    const float* __restrict__ Bp = B + (size_t)(2 * half) * LDB + nBase + l;

    v8f acc = {};
#pragma unroll
    for (int k0 = 0; k0 < K; k0 += 4) {
        v2f a, b;
        a[0] = Ap[k0];                         // A[m, k0+2*half]
        a[1] = Ap[k0 + 1];                     // A[m, k0+2*half+1]
        b[0] = Bp[(size_t)k0 * LDB];           // B[k0+2*half,   n]
        b[1] = Bp[(size_t)(k0 + 1) * LDB];     // B[k0+2*half+1, n]
        acc = __builtin_amdgcn_wmma_f32_16x16x4_f32(
                  false, a, false, b, (short)0, acc, false, false);
    }

#pragma unroll
    for (int r = 0; r < 8; ++r) {
        const int row = mBase + r + 8 * half;
        const int col = nBase + l;
        const size_t idx = (size_t)row * LDC + col;
        float v = acc[r];
        if constexpr (MODE == 1) {
            v += bias[col];
        } else if constexpr (MODE == 2) {
            v = pgm_softplus(v + bias[col]);
        } else if constexpr (MODE == 3) {
            v = C[idx] + scale * v;
            v = fminf(1.0f, fmaxf(-1.0f, v));
        }
        C[idx] = v;
    }
}

// ---------------------------------------------------------------------------
// Input build + lift (2->128) + plain LayerNorm. One block (128 thr) / token.
// ---------------------------------------------------------------------------
__global__ __launch_bounds__(128)
void pgm_lift_ln(const float* __restrict__ past,   // (B,K,N) flat == token idx
                 const float* __restrict__ gcoord, // (B,N)
                 const float* __restrict__ Wl,     // (2,128)
                 const float* __restrict__ bl,     // (128)
                 float* __restrict__ vout)         // (NTOK,128)
{
    const int t = blockIdx.x;
    const int j = threadIdx.x;
    const int m = t >> 9;          // sequence (b*K+k)
    const int n = t & 511;
    const int b = m / 3;           // K_STEPS == 3
    const float in0 = past[t];
    const float in1 = gcoord[b * NGRID + n];

    float v = in0 * Wl[j] + in1 * Wl[128 + j] + bl[j];

    __shared__ float red[128];
    red[j] = v;  __syncthreads();
    for (int s = 64; s > 0; s >>= 1) { if (j < s) red[j] += red[j + s]; __syncthreads(); }
    const float mean = red[0] * (1.0f / 128.0f);
    __syncthreads();
    const float d = v - mean;
    red[j] = d * d;  __syncthreads();
    for (int s = 64; s > 0; s >>= 1) { if (j < s) red[j] += red[j + s]; __syncthreads(); }
    const float var = red[0] * (1.0f / 128.0f);
    vout[(size_t)t * 128 + j] = d * rsqrtf(var + 1e-5f);
}

// ---------------------------------------------------------------------------
// LayerNorm(+affine)(+GELU) over 256 features. One block (256 thr) / token.
// X may alias Y (row value is register-held before the writing barrier).
// ---------------------------------------------------------------------------
__global__ __launch_bounds__(256)
void pgm_ln256(const float* __restrict__ X, float* __restrict__ Y,
               const float* __restrict__ g, const float* __restrict__ b,
               int do_gelu)
{
    const int t = blockIdx.x;
    const int j = threadIdx.x;
    const float x = X[(size_t)t * D_MODEL + j];

    __shared__ float red[256];
    red[j] = x;  __syncthreads();
    for (int s = 128; s > 0; s >>= 1) { if (j < s) red[j] += red[j + s]; __syncthreads(); }
    const float mean = red[0] * (1.0f / 256.0f);
    __syncthreads();
    const float d = x - mean;
    red[j] = d * d;  __syncthreads();
    for (int s = 128; s > 0; s >>= 1) { if (j < s) red[j] += red[j + s]; __syncthreads(); }
    const float var = red[0] * (1.0f / 256.0f);

    float y = d * rsqrtf(var + 1e-5f);
    y = y * g[j] + b[j];
    if (do_gelu) y = pgm_gelu(y);
    Y[(size_t)t * D_MODEL + j] = y;
}

// ---------------------------------------------------------------------------
// Causal depthwise conv (width 4) + bias + SiLU. x = xz[:, 0:512].
// One thread per (token, channel).
// ---------------------------------------------------------------------------
__global__ __launch_bounds__(256)
void pgm_conv_silu(const float* __restrict__ xz,   // (NTOK, 1024)
                   const float* __restrict__ cw,   // (512, 4)
                   const float* __restrict__ cb,   // (512)
                   float* __restrict__ xc)         // (NTOK, 512)
{
    const int gid = blockIdx.x * blockDim.x + threadIdx.x;
    const int c = gid & 511;
    const int t = gid >> 9;
    const int m = t >> 9;
    const int n = t & 511;
    float acc = cb[c];
#pragma unroll
    for (int j = 0; j < D_CONV; ++j) {
        const int nn = n - (D_CONV - 1) + j;
        if (nn >= 0)
            acc += cw[c * D_CONV + j] * xz[((size_t)(m * NGRID + nn)) * 1024 + c];
    }
    xc[(size_t)t * D_INNER + c] = pgm_silu(acc);
}

// ---------------------------------------------------------------------------
// Selective scan. One block (256 thr) per sequence; 512 serial steps.
// Each thread owns channels {tid, tid+256} with register state h[2][16].
// B,C broadcast through LDS. y overwrites xc (gated output), ready for
// the out_proj WMMA GEMM.
// ---------------------------------------------------------------------------
__global__ __launch_bounds__(256)
void pgm_scan(const float* __restrict__ xz,     // res = xz[:, 512:1024]
              float* __restrict__ xc,           // in: silu(conv(x)); out: y
              const float* __restrict__ delta,  // (NTOK, 512)
              const float* __restrict__ dbc,    // (NTOK, 48): [dt|B|C]
              const float* __restrict__ A_log,  // (512, 16)
              const float* __restrict__ Dp)     // (512)
{
    const int seq = blockIdx.x;
    const int tid = threadIdx.x;

    float h[2][16];
    float Arow[2][16];
    float Dv[2];
#pragma unroll
    for (int c2 = 0; c2 < 2; ++c2) {
        const int c = tid + 256 * c2;
        Dv[c2] = Dp[c];
#pragma unroll
        for (int s = 0; s < D_STATE; ++s) {
            Arow[c2][s] = -__expf(A_log[c * D_STATE + s]);
            h[c2][s] = 0.0f;
        }
    }

    __shared__ float sBC[32];   // [0:16)=B, [16:32)=C

    for (int n = 0; n < NGRID; ++n) {
        const int t = seq * NGRID + n;
        if (tid < 32) sBC[tid] = dbc[(size_t)t * 48 + DT_RANK + tid];
        __syncthreads();

#pragma unroll
        for (int c2 = 0; c2 < 2; ++c2) {
            const int c = tid + 256 * c2;
            const float dlt = delta[(size_t)t * D_INNER + c];
            const float xv  = xc[(size_t)t * D_INNER + c];
            float acc = Dv[c2] * xv;
#pragma unroll
            for (int s = 0; s < D_STATE; ++s) {
                const float hv = h[c2][s] * __expf(dlt * Arow[c2][s])
                               + dlt * sBC[s] * xv;
                h[c2][s] = hv;
                acc += hv * sBC[16 + s];
            }
            const float r = xz[(size_t)t * 1024 + D_INNER + c];
            xc[(size_t)t * D_INNER + c] = acc * pgm_silu(r);
        }
        __syncthreads();
    }
}

// ---------------------------------------------------------------------------
// Decode: gelu(z @ W_dec + b_dec) @ W_proj + b_proj -> scalar per token.
// One block (128 thr) per token.
// ---------------------------------------------------------------------------
__global__ __launch_bounds__(128)
void pgm_decode(const float* __restrict__ z,     // (NTOK, 256)
                const float* __restrict__ Wd,    // (256, 128)
                const float* __restrict__ bd,    // (128)
                const float* __restrict__ Wp,    // (128, 1)
                const float* __restrict__ bp,    // (1)
                float* __restrict__ gout)        // (NTOK)
{
    const int t = blockIdx.x;
    const int j = threadIdx.x;
    const float* zr = z + (size_t)t * D_MODEL;
    float acc = bd[j];
    for (int k = 0; k < D_MODEL; ++k)
        acc += zr[k] * Wd[k * 128 + j];
    acc = pgm_gelu(acc);

    __shared__ float red[128];
    red[j] = acc * Wp[j];  __syncthreads();
    for (int s = 64; s > 0; s >>= 1) { if (j < s) red[j] += red[j + s]; __syncthreads(); }
    if (j == 0) gout[t] = red[0] + bp[0];
}

// ---------------------------------------------------------------------------
// BDF-style combination + reg loss.  out[0:2048] = u_next, out[2048] = loss.
// ---------------------------------------------------------------------------
__global__ __launch_bounds__(256)
void pgm_finalize(const float* __restrict__ past,
                  const float* __restrict__ gout,
                  const float* __restrict__ decay,
                  float* __restrict__ out)
{
    const int gid = blockIdx.x * blockDim.x + threadIdx.x;
    const float dec = decay[0];
    if (gid < 2048) {
        const int b = gid >> 9;
        const int n = gid & 511;
        const float dw0 = dec * dec, dw1 = dec, dw2 = 1.0f;
        const float l0 =  2.0f / 11.0f, l1 = -9.0f / 11.0f, l2 = 18.0f / 11.0f;
        const float d2 =  6.0f / 11.0f;
        float u = l0 * dw0 * past[(b * 3 + 0) * NGRID + n]
                + l1 * dw1 * past[(b * 3 + 1) * NGRID + n]
                + l2 * dw2 * past[(b * 3 + 2) * NGRID + n]
                + DT_STEP * (d2 * dw2) * gout[(b * 3 + 2) * NGRID + n];
        out[gid] = u;
    }
    if (gid == 0) {
        const float e = dec - 0.95f;
        out[2048] = 1e-4f * e * e;
    }
}

// ---------------------------------------------------------------------------
// Host orchestration.
// Input order (setup dict order, params flattened in jax sorted-key order):
//  0 past(4,3,512) 1 grid(4,512,1) 2 W_dec 3 W_enc 4 W_lift 5 W_proj 6 b_dec
//  7 b_enc 8 b_lift 9 b_proj 10 beta_enc 11 decay 12 g_enc
//  13 L.A_log 14 L.D 15 L.conv_b 16 L.conv_w 17 L.dt_b 18 L.dt_w
//  19 L.in_proj 20 L.out_proj 21 L.x_proj 22 ln_b 23 ln_g
// ---------------------------------------------------------------------------
extern "C" void kernel_launch(void* const* d_in, const int* in_sizes, int n_in,
                              void* d_out, int out_size, void* d_ws, size_t ws_size,
                              hipStream_t stream)
{
    (void)in_sizes; (void)n_in; (void)out_size; (void)ws_size;

    const float* past    = (const float*)d_in[0];
    const float* gcoord  = (const float*)d_in[1];
    const float* W_dec   = (const float*)d_in[2];
    const float* W_enc   = (const float*)d_in[3];
    const float* W_lift  = (const float*)d_in[4];
    const float* W_proj  = (const float*)d_in[5];
    const float* b_dec   = (const float*)d_in[6];
    const float* b_enc   = (const float*)d_in[7];
    const float* b_lift  = (const float*)d_in[8];
    const float* b_proj  = (const float*)d_in[9];
    const float* beta_e  = (const float*)d_in[10];
    const float* decay   = (const float*)d_in[11];
    const float* g_enc   = (const float*)d_in[12];
    const float* A_log   = (const float*)d_in[13];
    const float* Dparam  = (const float*)d_in[14];
    const float* conv_b  = (const float*)d_in[15];
    const float* conv_w  = (const float*)d_in[16];
    const float* dt_b    = (const float*)d_in[17];
    const float* dt_w    = (const float*)d_in[18];
    const float* Win     = (const float*)d_in[19];
    const float* Wout    = (const float*)d_in[20];
    const float* Wx      = (const float*)d_in[21];
    const float* ln_b    = (const float*)d_in[22];
    const float* ln_g    = (const float*)d_in[23];

    float* out = (float*)d_out;

    // Workspace layout (floats). Total ~16.0M floats (~64.1 MB) — L2-resident.
    float* ws = (float*)d_ws;
    float* z     = ws;                              // NTOK*256
    float* zn    = z     + (size_t)NTOK * 256;      // NTOK*256 (also 'v' lift buf)
    float* xz    = zn    + (size_t)NTOK * 256;      // NTOK*1024
    float* xc    = xz    + (size_t)NTOK * 1024;     // NTOK*512
    float* dbc   = xc    + (size_t)NTOK * 512;      // NTOK*48
    float* delta = dbc   + (size_t)NTOK * 48;       // NTOK*512
    float* gbuf  = delta + (size_t)NTOK * 512;      // NTOK

    const dim3 blk(256);
    const int MG = NTOK / 128;                      // 48 M-tiles

    // ---- encoder --------------------------------------------------------
    pgm_lift_ln<<<NTOK, 128, 0, stream>>>(past, gcoord, W_lift, b_lift, zn);
    // z = v @ W_enc + b_enc   (K=128, N=256)
    pgm_gemm16<128, 128, 256, 256, 1>
        <<<dim3(256 / 16, MG), blk, 0, stream>>>(zn, W_enc, z, b_enc, 0.0f);
    // in-place LN affine + GELU
    pgm_ln256<<<NTOK, 256, 0, stream>>>(z, z, g_enc, beta_e, 1);

    // ---- mamba layers ---------------------------------------------------
    for (int i = 0; i < NLAYERS; ++i) {
        const float* Win_i  = Win    + (size_t)i * D_MODEL * (2 * D_INNER);
        const float* Wout_i = Wout   + (size_t)i * D_INNER * D_MODEL;
        const float* Wx_i   = Wx     + (size_t)i * D_INNER * 48;
        const float* dtw_i  = dt_w   + (size_t)i * DT_RANK * D_INNER;
        const float* dtb_i  = dt_b   + (size_t)i * D_INNER;
        const float* cw_i   = conv_w + (size_t)i * D_INNER * D_CONV;
        const float* cb_i   = conv_b + (size_t)i * D_INNER;
        const float* Al_i   = A_log  + (size_t)i * D_INNER * D_STATE;
        const float* D_i    = Dparam + (size_t)i * D_INNER;
        const float* lg_i   = ln_g   + (size_t)i * D_MODEL;
        const float* lb_i   = ln_b   + (size_t)i * D_MODEL;
        const float scale   = (i == 0) ? 1.0f : 0.1f;

        // zn = LN(z)*g+b
        pgm_ln256<<<NTOK, 256, 0, stream>>>(z, zn, lg_i, lb_i, 0);
        // xz = zn @ in_proj     (K=256, N=1024)
        pgm_gemm16<256, 256, 1024, 1024, 0>
            <<<dim3(1024 / 16, MG), blk, 0, stream>>>(zn, Win_i, xz, nullptr, 0.0f);
        // xc = silu(causal_conv(x) + cb)
        pgm_conv_silu<<<(NTOK * D_INNER) / 256, 256, 0, stream>>>(xz, cw_i, cb_i, xc);
        // dbc = xc @ x_proj     (K=512, N=48)
        pgm_gemm16<512, 512, 48, 48, 0>
            <<<dim3(48 / 16, MG), blk, 0, stream>>>(xc, Wx_i, dbc, nullptr, 0.0f);
        // delta = softplus(dbc[:, :16] @ dt_w + dt_b)   (K=16, N=512)
        pgm_gemm16<16, 48, 512, 512, 2>
            <<<dim3(512 / 16, MG), blk, 0, stream>>>(dbc, dtw_i, delta, dtb_i, 0.0f);
        // serial selective scan + gating; y overwrites xc
        pgm_scan<<<NSEQ, 256, 0, stream>>>(xz, xc, delta, dbc, Al_i, D_i);
        // z = clip(z + scale * (y @ out_proj), -1, 1)   (K=512, N=256)
        pgm_gemm16<512, 512, 256, 256, 3>
            <<<dim3(256 / 16, MG), blk, 0, stream>>>(xc, Wout_i, z, nullptr, scale);
    }

    // ---- decoder + time step -------------------------------------------
    pgm_decode<<<NTOK, 128, 0, stream>>>(z, W_dec, b_dec, W_proj, b_proj, gbuf);
    pgm_finalize<<<(2048 + 255) / 256, 256, 0, stream>>>(past, gbuf, decay, out);
}